// HashEmbedder_25486335934782
// MI455X (gfx1250) — compile-verified
//
#include <hip/hip_runtime.h>
#include <hip/hip_bf16.h>
#include <stdint.h>

typedef float v2f __attribute__((ext_vector_type(2)));

#define NLVL   16
#define LOG2T  19
#define TBL    (1u << LOG2T)
#define HMASK  (TBL - 1u)

// res_l = floor(16 * 32^(l/15))  (16 -> 512 geometric)
__constant__ float c_res[NLVL] = {
    16.f, 20.f, 25.f, 32.f, 40.f, 50.f, 64.f, 80.f,
    101.f, 128.f, 161.f, 203.f, 256.f, 322.f, 406.f, 512.f
};

__global__ __launch_bounds__(256)
void hashgrid_fwd(const float* __restrict__ x,     // [B,3]
                  const float* __restrict__ emb,   // [16, 2^19, 2]
                  float* __restrict__ out,         // [B,32]
                  int B)
{
    __shared__ float sx[48];                       // 16 points * 3 coords
    const int t = threadIdx.x;
    const long long ptBase = (long long)blockIdx.x * 16;

    // ---- Stage this block's 16 points into LDS via async global->LDS copy
    // (gfx1250 ASYNCcnt path). 48 lanes each move one dword.
    if (t < 48) {
        long long g = ptBase * 3 + t;
        if (g < (long long)B * 3) {
            uint32_t lds_off = (uint32_t)(uintptr_t)(&sx[t]);
            uint64_t gaddr   = (uint64_t)(uintptr_t)(x + g);
            asm volatile("global_load_async_to_lds_b32 %0, %1, off"
                         :: "v"(lds_off), "v"(gaddr) : "memory");
        }
    }
    // Prefetch next block's coordinates while the async copy is in flight.
    if (t == 0 && (ptBase + 16) < (long long)B) {
        __builtin_prefetch(x + (ptBase + 16) * 3, 0, 0);
    }
    asm volatile("s_wait_asynccnt 0x0" ::: "memory");
    __syncthreads();

    const int bl = t >> 4;          // point within block: 0..15
    const int l  = t & 15;          // level: 0..15  (fastest -> coalesced stores)
    const long long b = ptBase + bl;
    if (b >= B) return;

    float px = sx[bl * 3 + 0];
    float py = sx[bl * 3 + 1];
    float pz = sx[bl * 3 + 2];
    px = fminf(fmaxf(px, -1.f), 1.f);
    py = fminf(fmaxf(py, -1.f), 1.f);
    pz = fminf(fmaxf(pz, -1.f), 1.f);

    const float res = c_res[l];
    const float gs  = 2.0f / res;   // grid_size = (BOX_MAX-BOX_MIN)/res

    float fx = floorf((px + 1.0f) / gs);
    float fy = floorf((py + 1.0f) / gs);
    float fz = floorf((pz + 1.0f) / gs);
    int ix = (int)fx, iy = (int)fy, iz = (int)fz;

    // w = (xc - (bl*gs + BOX_MIN)) / gs
    float wx = (px - (fx * gs - 1.0f)) / gs;
    float wy = (py - (fy * gs - 1.0f)) / gs;
    float wz = (pz - (fz * gs - 1.0f)) / gs;

    // int32-wraparound hash pieces (uint arithmetic == int32 wrap bits)
    unsigned hx0 = (unsigned)ix * 73856093u;
    unsigned hx1 = hx0 + 73856093u;
    unsigned hy0 = (unsigned)iy * 19349663u;
    unsigned hy1 = hy0 + 19349663u;
    unsigned hz0 = (unsigned)iz * 83492791u;
    unsigned hz1 = hz0 + 83492791u;

    // corner c = 4*i + 2*j + k  (matches OFFSETS ordering)
    unsigned h0 = (hx0 ^ hy0 ^ hz0) & HMASK;
    unsigned h1 = (hx0 ^ hy0 ^ hz1) & HMASK;
    unsigned h2 = (hx0 ^ hy1 ^ hz0) & HMASK;
    unsigned h3 = (hx0 ^ hy1 ^ hz1) & HMASK;
    unsigned h4 = (hx1 ^ hy0 ^ hz0) & HMASK;
    unsigned h5 = (hx1 ^ hy0 ^ hz1) & HMASK;
    unsigned h6 = (hx1 ^ hy1 ^ hz0) & HMASK;
    unsigned h7 = (hx1 ^ hy1 ^ hz1) & HMASK;

    // 8 independent 8B gathers; tables (64MB total) are L2-resident, keep
    // default RT policy so they stay hot. 32-bit offsets -> SADDR+voffset form.
    const char* base = (const char*)emb;
    uint32_t lofs = ((uint32_t)l << LOG2T);
    v2f e0 = *(const v2f*)(base + (uint64_t)((lofs + h0) * 8u));
    v2f e1 = *(const v2f*)(base + (uint64_t)((lofs + h1) * 8u));
    v2f e2 = *(const v2f*)(base + (uint64_t)((lofs + h2) * 8u));
    v2f e3 = *(const v2f*)(base + (uint64_t)((lofs + h3) * 8u));
    v2f e4 = *(const v2f*)(base + (uint64_t)((lofs + h4) * 8u));
    v2f e5 = *(const v2f*)(base + (uint64_t)((lofs + h5) * 8u));
    v2f e6 = *(const v2f*)(base + (uint64_t)((lofs + h6) * 8u));
    v2f e7 = *(const v2f*)(base + (uint64_t)((lofs + h7) * 8u));

    float wx0 = 1.0f - wx, wy0 = 1.0f - wy, wz0 = 1.0f - wz;
    float c0 = wx0 * wy0 * wz0;
    float c1 = wx0 * wy0 * wz;
    float c2 = wx0 * wy  * wz0;
    float c3 = wx0 * wy  * wz;
    float c4 = wx  * wy0 * wz0;
    float c5 = wx  * wy0 * wz;
    float c6 = wx  * wy  * wz0;
    float c7 = wx  * wy  * wz;

    float ox = c0 * e0.x;
    float oy = c0 * e0.y;
    ox = fmaf(c1, e1.x, ox);  oy = fmaf(c1, e1.y, oy);
    ox = fmaf(c2, e2.x, ox);  oy = fmaf(c2, e2.y, oy);
    ox = fmaf(c3, e3.x, ox);  oy = fmaf(c3, e3.y, oy);
    ox = fmaf(c4, e4.x, ox);  oy = fmaf(c4, e4.y, oy);
    ox = fmaf(c5, e5.x, ox);  oy = fmaf(c5, e5.y, oy);
    ox = fmaf(c6, e6.x, ox);  oy = fmaf(c6, e6.y, oy);
    ox = fmaf(c7, e7.x, ox);  oy = fmaf(c7, e7.y, oy);

    // 128MB output stream: non-temporal so it doesn't evict the L2-resident
    // hash tables. Wave32 writes 256 contiguous bytes (l is the fast index).
    v2f o = { ox, oy };
    __builtin_nontemporal_store(o, (v2f*)out + (b * 16 + l));
}

extern "C" void kernel_launch(void* const* d_in, const int* in_sizes, int n_in,
                              void* d_out, int out_size, void* d_ws, size_t ws_size,
                              hipStream_t stream) {
    const float* x   = (const float*)d_in[0];   // [B,3] f32
    const float* emb = (const float*)d_in[1];   // [16, 2^19, 2] f32
    float* out = (float*)d_out;                 // [B,32] f32
    int B = in_sizes[0] / 3;
    int blocks = (B + 15) / 16;                 // 16 points per 256-thread block
    hashgrid_fwd<<<blocks, 256, 0, stream>>>(x, emb, out, B);
}